// PMVAE_34961033789789
// MI455X (gfx1250) — compile-verified
//
#include <hip/hip_runtime.h>
#include <cstddef>

// ---------------- problem dimensions (fixed by the reference) ----------------
#define BATCH 8192
#define GENES 5000
#define H1    4128   // NM*HID
#define H2    2064   // 2*NM*LAT
#define NLAT  1032   // NM*LAT
#define HD    4128
#define BN_EPS 1e-3f

typedef float v2f __attribute__((ext_vector_type(2)));
typedef float v4f __attribute__((ext_vector_type(4)));
typedef float v8f __attribute__((ext_vector_type(8)));
typedef int   i32x4 __attribute__((vector_size(16)));   // matches builtin param type

// ---------------- GEMM tiling ----------------
#define TBM 128
#define TBN 128
#define TBK 16
#define LDT 20   // padded LDS row stride (floats): conflict-free, 16B-aligned rows

#define ASYNC_GBL(p) ((__attribute__((address_space(1))) i32x4*)(p))
#define ASYNC_LDS(p) ((__attribute__((address_space(3))) i32x4*)(p))

__device__ __forceinline__ void wait_async_zero() {
#if __has_builtin(__builtin_amdgcn_s_wait_asynccnt)
    __builtin_amdgcn_s_wait_asynccnt(0);
#else
    asm volatile("s_wait_asynccnt 0x0" ::: "memory");
#endif
}

// C(M,N) = A(M,K) @ (W(N,K) * Wmask(N,K))^T + bias
// Split store: col < Ns -> out0 (stride Ns), else -> out1 (stride N-Ns).
__global__ __launch_bounds__(256)
void gemm_nt_masked(const float* __restrict__ A,
                    const float* __restrict__ W,
                    const float* __restrict__ Wmask,
                    const float* __restrict__ bias,
                    float* __restrict__ out0,
                    float* __restrict__ out1,
                    int M, int N, int K, int Ns)
{
    __shared__ float As[2][TBM * LDT];
    __shared__ float Bs[2][TBM * LDT];

    const int tid  = threadIdx.x;
    const int lane = tid & 31;
    const int wave = tid >> 5;            // 0..7
    const int wm   = (wave >> 2) * 64;    // 2 waves along M
    const int wn   = (wave & 3) * 32;     // 4 waves along N
    const int bm   = blockIdx.y * TBM;
    const int bn   = blockIdx.x * TBN;

    // tile loader mapping: 256 threads x 2 rows, one float4 (4 k's) each
    const int lk = (tid & 3) * 4;         // 0,4,8,12
    const int lr = tid >> 2;              // 0..63

    v8f acc[4][2];
#pragma unroll
    for (int i = 0; i < 4; ++i)
#pragma unroll
        for (int j = 0; j < 2; ++j)
            acc[i][j] = (v8f){0.f,0.f,0.f,0.f,0.f,0.f,0.f,0.f};

    auto load_tile = [&](int kbase, int buf) {
#pragma unroll
        for (int h = 0; h < 2; ++h) {
            const int r  = lr + h * 64;
            const int gk = kbase + lk;            // K % 4 == 0 for all GEMMs
            const int gr = bm + r;
            float* adst = &As[buf][r * LDT + lk];
            if (gr < M && gk < K) {
                // CDNA5 async DMA: global -> LDS without touching VGPRs
#if __has_builtin(__builtin_amdgcn_global_load_async_to_lds_b128)
                __builtin_amdgcn_global_load_async_to_lds_b128(
                    ASYNC_GBL(A + (size_t)gr * K + gk), ASYNC_LDS(adst), 0, 0);
#else
                *(v4f*)adst = *(const v4f*)(A + (size_t)gr * K + gk);
#endif
            } else {
                *(v4f*)adst = (v4f){0.f, 0.f, 0.f, 0.f};
            }

            // weight tile: synchronous load so the pathway mask can be fused
            v4f wv = {0.f, 0.f, 0.f, 0.f};
            const int gn = bn + r;
            if (gn < N && gk < K) {
                wv = *(const v4f*)(W + (size_t)gn * K + gk);
                const v4f mv = *(const v4f*)(Wmask + (size_t)gn * K + gk);
                wv *= mv;                          // fuse mask into weight tile
                if (gk + 2 * TBK < K)
                    __builtin_prefetch(W + (size_t)gn * K + gk + 2 * TBK, 0, 1);
            }
            *(v4f*)(&Bs[buf][r * LDT + lk]) = wv;
        }
    };

    const int nkt = (K + TBK - 1) / TBK;
    load_tile(0, 0);
    wait_async_zero();
    __syncthreads();

    const int mrow = lane & 15;        // A/B fragment row within 16
    const int khal = (lane >> 4) * 2;  // K offset selected by lane half

    for (int kt = 0; kt < nkt; ++kt) {
        const int cur = kt & 1;
        if (kt + 1 < nkt) load_tile((kt + 1) * TBK, cur ^ 1);

        const float* Ab = &As[cur][0];
        const float* Bb = &Bs[cur][0];

        auto kstep = [&](int kk) {
            const int kcol = kk + khal;
            v2f afr[4], bfr[2];
#pragma unroll
            for (int i = 0; i < 4; ++i)
                afr[i] = *(const v2f*)(Ab + (wm + i * 16 + mrow) * LDT + kcol);
#pragma unroll
            for (int j = 0; j < 2; ++j)
                bfr[j] = *(const v2f*)(Bb + (wn + j * 16 + mrow) * LDT + kcol);
#pragma unroll
            for (int i = 0; i < 4; ++i)
#pragma unroll
                for (int j = 0; j < 2; ++j)
                    acc[i][j] = __builtin_amdgcn_wmma_f32_16x16x4_f32(
                        false, afr[i], false, bfr[j],
                        (short)0, acc[i][j], false, false);
        };

        int kmax = K - kt * TBK; if (kmax > TBK) kmax = TBK;   // uniform bound
        if (kmax == TBK) {
#pragma unroll
            for (int kk = 0; kk < TBK; kk += 4) kstep(kk);     // 32 WMMA straight-line
        } else {
            for (int kk = 0; kk < kmax; kk += 4) kstep(kk);    // K-tail (GEMM1 and decoder GEMM)
        }

        wait_async_zero();
        __syncthreads();
    }

    // epilogue: lane l -> n = l&15 ; vgpr r -> m = r + 8*(l>>4)
    const int half = lane >> 4;
    const int nloc = lane & 15;
#pragma unroll
    for (int i = 0; i < 4; ++i) {
#pragma unroll
        for (int j = 0; j < 2; ++j) {
            const int gn = bn + wn + j * 16 + nloc;
            if (gn >= N) continue;
            const float bv = bias ? bias[gn] : 0.0f;
#pragma unroll
            for (int r = 0; r < 8; ++r) {
                const int gm = bm + wm + i * 16 + half * 8 + r;
                if (gm >= M) continue;
                const float val = acc[i][j][r] + bv;
                if (gn < Ns) out0[(size_t)gm * Ns + gn] = val;
                else         out1[(size_t)gm * (N - Ns) + (gn - Ns)] = val;
            }
        }
    }
}

// Per-feature batch mean / rsqrt(var+eps) over M rows (deterministic tree).
__global__ __launch_bounds__(256)
void bn_stats(const float* __restrict__ X, int M, int N,
              float* __restrict__ mean, float* __restrict__ rstd)
{
    __shared__ float s_sum[8][32];
    __shared__ float s_sq[8][32];
    const int c   = threadIdx.x & 31;
    const int rg  = threadIdx.x >> 5;
    const int col = blockIdx.x * 32 + c;
    float s = 0.f, q = 0.f;
    if (col < N) {
        for (int r = rg; r < M; r += 8) {
            const float v = X[(size_t)r * N + col];
            s += v; q += v * v;
        }
    }
    s_sum[rg][c] = s;
    s_sq[rg][c]  = q;
    __syncthreads();
    if (rg == 0 && col < N) {
#pragma unroll
        for (int k = 1; k < 8; ++k) { s += s_sum[k][c]; q += s_sq[k][c]; }
        const float m = s / (float)M;
        const float v = q / (float)M - m * m;
        mean[col] = m;
        rstd[col] = rsqrtf(v + BN_EPS);
    }
}

// y = g*(x-mean)*rstd + b  (+ optional ELU). src may alias dst.
__global__ __launch_bounds__(256)
void bn_apply(const float* __restrict__ src, float* __restrict__ dst,
              const float* __restrict__ mean, const float* __restrict__ rstd,
              const float* __restrict__ g, const float* __restrict__ be,
              unsigned M, unsigned N, int do_elu)
{
    const unsigned total = M * N;
    for (unsigned idx = blockIdx.x * 256u + threadIdx.x; idx < total;
         idx += gridDim.x * 256u) {
        const unsigned n = idx % N;
        float y = g[n] * (src[idx] - mean[n]) * rstd[n] + be[n];
        if (do_elu) y = (y > 0.f) ? y : (expf(y) - 1.f);
        dst[idx] = y;
    }
}

// z = mu + exp(logvar/2) * eps
__global__ __launch_bounds__(256)
void reparam(const float* __restrict__ mu, const float* __restrict__ lv,
             const float* __restrict__ eps, float* __restrict__ z, unsigned total)
{
    for (unsigned idx = blockIdx.x * 256u + threadIdx.x; idx < total;
         idx += gridDim.x * 256u)
        z[idx] = mu[idx] + expf(0.5f * lv[idx]) * eps[idx];
}

extern "C" void kernel_launch(void* const* d_in, const int* in_sizes, int n_in,
                              void* d_out, int out_size, void* d_ws, size_t ws_size,
                              hipStream_t stream)
{
    (void)in_sizes; (void)n_in; (void)out_size; (void)ws_size;
    const float* x     = (const float*)d_in[0];
    const float* eps   = (const float*)d_in[1];
    const float* mask1 = (const float*)d_in[2];
    const float* mask2 = (const float*)d_in[3];
    const float* maskd = (const float*)d_in[4];
    const float* maskm = (const float*)d_in[5];
    const float* W1    = (const float*)d_in[6];
    const float* b1    = (const float*)d_in[7];
    const float* g1    = (const float*)d_in[8];
    const float* be1   = (const float*)d_in[9];
    const float* W2    = (const float*)d_in[10];
    const float* b2    = (const float*)d_in[11];
    const float* g2    = (const float*)d_in[12];
    const float* be2   = (const float*)d_in[13];
    const float* Wd    = (const float*)d_in[14];
    const float* bd    = (const float*)d_in[15];
    const float* gd    = (const float*)d_in[16];
    const float* bed   = (const float*)d_in[17];
    const float* Wm    = (const float*)d_in[18];

    float* out   = (float*)d_out;
    float* recon = out;                                  // 8192 x 5000
    float* zbuf  = recon + (size_t)BATCH * GENES;        // 8192 x 1032
    float* mod   = zbuf  + (size_t)BATCH * NLAT;         // 8192 x 4128
    float* mu    = mod   + (size_t)BATCH * HD;           // 8192 x 1032
    float* lv    = mu    + (size_t)BATCH * NLAT;         // 8192 x 1032

    // tiny workspace: BN statistics only
    float* ws     = (float*)d_ws;
    float* mean1  = ws;              float* rstd1  = mean1  + H1;
    float* meanMu = rstd1  + H1;     float* rstdMu = meanMu + NLAT;
    float* meanLv = rstdMu + NLAT;   float* rstdLv = meanLv + NLAT;
    float* meanD  = rstdLv + NLAT;   float* rstdD  = meanD  + HD;

    float* h = recon;  // stage pre-BN encoder hidden in the (written-last) recon slot

    const dim3 blk(256);
    const unsigned MB = BATCH / TBM;  // 64

    // encoder L1: h = x @ (W1*mask1)^T + b1 ; BN ; ELU (in place)
    gemm_nt_masked<<<dim3((H1 + TBN - 1) / TBN, MB), blk, 0, stream>>>(
        x, W1, mask1, b1, h, nullptr, BATCH, H1, GENES, H1);
    bn_stats<<<dim3((H1 + 31) / 32), blk, 0, stream>>>(h, BATCH, H1, mean1, rstd1);
    bn_apply<<<dim3(8192), blk, 0, stream>>>(h, h, mean1, rstd1, g1, be1, BATCH, H1, 1);

    // encoder L2: p = h @ (W2*mask2)^T + b2, split-stored into mu/logvar slots
    gemm_nt_masked<<<dim3((H2 + TBN - 1) / TBN, MB), blk, 0, stream>>>(
        h, W2, mask2, b2, mu, lv, BATCH, H2, H1, NLAT);
    bn_stats<<<dim3((NLAT + 31) / 32), blk, 0, stream>>>(mu, BATCH, NLAT, meanMu, rstdMu);
    bn_apply<<<dim3(4096), blk, 0, stream>>>(mu, mu, meanMu, rstdMu, g2, be2, BATCH, NLAT, 0);
    bn_stats<<<dim3((NLAT + 31) / 32), blk, 0, stream>>>(lv, BATCH, NLAT, meanLv, rstdLv);
    bn_apply<<<dim3(4096), blk, 0, stream>>>(lv, lv, meanLv, rstdLv,
                                             g2 + NLAT, be2 + NLAT, BATCH, NLAT, 0);
    reparam<<<dim3(4096), blk, 0, stream>>>(mu, lv, eps, zbuf,
                                            (unsigned)((size_t)BATCH * NLAT));

    // decoder: d = z @ (Wd*maskd)^T + bd ; BN ; ELU (in place in module slot)
    gemm_nt_masked<<<dim3((HD + TBN - 1) / TBN, MB), blk, 0, stream>>>(
        zbuf, Wd, maskd, bd, mod, nullptr, BATCH, HD, NLAT, HD);
    bn_stats<<<dim3((HD + 31) / 32), blk, 0, stream>>>(mod, BATCH, HD, meanD, rstdD);
    bn_apply<<<dim3(8192), blk, 0, stream>>>(mod, mod, meanD, rstdD, gd, bed, BATCH, HD, 1);

    // merger: recon = module_outputs @ (Wm*maskm)^T   (overwrites h staging)
    gemm_nt_masked<<<dim3((GENES + TBN - 1) / TBN, MB), blk, 0, stream>>>(
        mod, Wm, maskm, nullptr, recon, nullptr, BATCH, GENES, HD, GENES);
}